// GPTBigCodeAttention_53249004536100
// MI455X (gfx1250) — compile-verified
//
#include <hip/hip_runtime.h>
#include <stdint.h>

// Problem constants (from reference)
#define SEQ   32
#define HID   6144
#define NHEAD 48
#define DHEAD 128
#define BSZ   16
#define MAXB  256

typedef float v2f __attribute__((ext_vector_type(2)));
typedef float v8f __attribute__((ext_vector_type(8)));

// D = A(16x4 f32) * B(4x16 f32) + C(16x16 f32)
// A frag: lane holds M = lane&15, K = {off, off+1}, off = 2*(lane>>4)
// B frag: lane holds N = lane&15, K = {off, off+1}
// C frag: reg i holds row M = i + 8*(lane>>4), col N = lane&15
__device__ __forceinline__ v8f wmma4(v2f a, v2f b, v8f c) {
  return __builtin_amdgcn_wmma_f32_16x16x4_f32(false, a, false, b, (short)0, c,
                                               false, false);
}

// CDNA5 async global->LDS copy (ASYNCcnt-tracked), 16 bytes per enabled lane.
__device__ __forceinline__ void async_g2l_b128(void* lds_dst, const void* gsrc) {
  uint32_t ldsa = (uint32_t)(uintptr_t)lds_dst;  // flat LDS addr truncates to offset
  asm volatile("global_load_async_to_lds_b128 %0, %1, off"
               :: "v"(ldsa), "v"(gsrc)
               : "memory");
}
__device__ __forceinline__ void wait_asynccnt0() {
  asm volatile("s_wait_asynccnt 0x0" ::: "memory");
}

// ---------------------------------------------------------------------------
// Skinny GEMM: C(32 x N) = A(32 x K=6144) * B(K x N) + bias, fp32 WMMA 16x16x4.
// One 16-wide N tile per block; 4 waves split K; LDS reduction (deterministic).
// ---------------------------------------------------------------------------
__global__ __launch_bounds__(128)
void gemm32_wmma(const float* __restrict__ A, const float* __restrict__ B,
                 const float* __restrict__ bias, float* __restrict__ C, int N) {
  constexpr int K = HID;
  __shared__ float red[4][32][16];

  const int tid  = threadIdx.x;
  const int lane = tid & 31;
  const int wv   = tid >> 5;            // 0..3 : K segment
  const int m    = lane & 15;
  const int off  = (lane >> 4) << 1;    // 0 or 2
  const int n    = blockIdx.x * 16 + m; // output column

  constexpr int kseg = K / 4;           // 1536, multiple of 4
  const int k0 = wv * kseg;

  const float* ap0 = A + (size_t)m * K + k0 + off;
  const float* ap1 = A + (size_t)(m + 16) * K + k0 + off;
  const float* bp  = B + (size_t)(k0 + off) * N + n;      // row k+off
  const float* bq  = bp + N;                              // row k+off+1
  const float* bpf = bp + (size_t)128 * N;                // prefetch ~128 rows ahead
  const size_t bstep = (size_t)4 * N;

  v8f acc0 = {};
  v8f acc1 = {};

#pragma unroll 8
  for (int k = 0; k < kseg; k += 4) {
    v2f a0 = *(const v2f*)(ap0 + k);
    v2f a1 = *(const v2f*)(ap1 + k);
    v2f b;
    b.x = bp[0];
    b.y = bq[0];
    __builtin_prefetch(bpf, 0, 3);
    bp  += bstep;
    bq  += bstep;
    bpf += bstep;
    acc0 = wmma4(a0, b, acc0);
    acc1 = wmma4(a1, b, acc1);
  }

#pragma unroll
  for (int i = 0; i < 8; ++i) {
    red[wv][lane][i]     = acc0[i];
    red[wv][lane][i + 8] = acc1[i];
  }
  __syncthreads();

  if (wv == 0) {
    const float bv  = bias[n];
    const int rbase = (lane >> 4) << 3;   // 0 or 8
#pragma unroll
    for (int i = 0; i < 8; ++i) {
      float s0 = red[0][lane][i]   + red[1][lane][i]
               + red[2][lane][i]   + red[3][lane][i];
      float s1 = red[0][lane][i+8] + red[1][lane][i+8]
               + red[2][lane][i+8] + red[3][lane][i+8];
      int r = i + rbase;
      C[(size_t)r        * N + n] = s0 + bv;
      C[(size_t)(r + 16) * N + n] = s1 + bv;
    }
  }
}

// ---------------------------------------------------------------------------
// Paged MQA flash-decode, fp32 WMMA, async global->LDS staging.
// grid.x = seq (32). block = 96 threads = 3 waves, one 16-head tile per wave.
// All waves share the staged K/V tile (MQA: one KV head for 48 Q heads).
// Self token (k_new/v_new) is folded in as position == ctx_len.
// Positions > ctx read a valid-but-ignored cache row; their score columns are
// forced to -3e38 before softmax so p underflows to exactly 0.
// ---------------------------------------------------------------------------
#define KSTR 132   // padded LDS row stride (floats) for K/V tiles
#define PSTR 18    // padded LDS row stride for P tile

__global__ __launch_bounds__(96)
void paged_mqa_wmma(const float* __restrict__ qbuf,
                    const float* __restrict__ kvbuf,
                    const float* __restrict__ kcache,
                    const float* __restrict__ vcache,
                    const int*   __restrict__ btab,
                    const int*   __restrict__ ctxlen,
                    float* __restrict__ out) {
  __shared__ float Kl[16 * KSTR];
  __shared__ float Vl[16 * KSTR];
  __shared__ float Pl[3][16 * PSTR];

  const int tid  = threadIdx.x;
  const int lane = tid & 31;
  const int wv   = tid >> 5;           // head tile 0..2
  const int s    = blockIdx.x;         // sequence
  const int m    = lane & 15;
  const int off  = (lane >> 4) << 1;
  const int rbase = (lane >> 4) << 3;
  const int ctx  = ctxlen[s];
  const float scale = 0.08838834764831845f;  // 1/sqrt(128)

  // Preload q fragments for this wave's 16 heads (A-layout, 64 VGPRs).
  v2f qa[32];
  const float* qrow =
      qbuf + (size_t)s * (NHEAD * DHEAD) + (size_t)(wv * 16 + m) * DHEAD;
#pragma unroll
  for (int c = 0; c < 32; ++c)
    qa[c] = *(const v2f*)(qrow + 4 * c + off);

  v8f acc[8];
  v8f vzero = {};
#pragma unroll
  for (int ct = 0; ct < 8; ++ct) acc[ct] = vzero;

  float mrun[8], lrun[8];
#pragma unroll
  for (int i = 0; i < 8; ++i) { mrun[i] = -3.0e38f; lrun[i] = 0.0f; }

  const int nsteps = (ctx + 1 + 15) >> 4;  // ctx tokens + self token
  const float* knew = kvbuf + (size_t)s * 256;          // k_new row (128 f32)
  const float* vnew = knew + DHEAD;                     // v_new row

  for (int jb = 0; jb < nsteps; ++jb) {
    // ---- async staging of one 16(pos) x 128(dim) K/V tile into LDS ----
    const int blk = (jb * BSZ < ctx) ? btab[s * MAXB + jb] : 0;
    for (int e = tid; e < 512; e += 96) {       // 512 float4 per tensor
      int pos = e >> 5;
      int d4  = e & 31;
      int gp  = jb * BSZ + pos;
      const float* ks = kcache + ((size_t)blk * BSZ + pos) * DHEAD + 4 * d4;
      const float* vs = vcache + ((size_t)blk * BSZ + pos) * DHEAD + 4 * d4;
      if (gp == ctx) {                          // self token: k_new / v_new
        ks = knew + 4 * d4;
        vs = vnew + 4 * d4;
      }
      async_g2l_b128(&Kl[pos * KSTR + 4 * d4], ks);
      async_g2l_b128(&Vl[pos * KSTR + 4 * d4], vs);
    }
    wait_asynccnt0();
    __syncthreads();

    // ---- scores(16 heads x 16 pos) = q_tile @ K^T, K-dim = 128 ----
    v8f sc = vzero;
#pragma unroll
    for (int c = 0; c < 32; ++c) {
      v2f b = *(const v2f*)&Kl[m * KSTR + 4 * c + off];  // B[k][n]=K[n][k]
      sc = wmma4(qa[c], b, sc);
    }

    // ---- online softmax update ----
    const int  gpc   = jb * BSZ + m;      // position of this lane's column
    const bool valid = (gpc <= ctx);
    float p[8], f[8];
#pragma unroll
    for (int i = 0; i < 8; ++i) {
      float sv = valid ? sc[i] * scale : -3.0e38f;
      float bm = sv;                       // row max over 16-lane half
      bm = fmaxf(bm, __shfl_xor(bm, 1, 32));
      bm = fmaxf(bm, __shfl_xor(bm, 2, 32));
      bm = fmaxf(bm, __shfl_xor(bm, 4, 32));
      bm = fmaxf(bm, __shfl_xor(bm, 8, 32));
      float nm = fmaxf(mrun[i], bm);
      f[i] = __expf(mrun[i] - nm);
      p[i] = __expf(sv - nm);
      mrun[i] = nm;
      float rs = p[i];                     // row sum over 16-lane half
      rs += __shfl_xor(rs, 1, 32);
      rs += __shfl_xor(rs, 2, 32);
      rs += __shfl_xor(rs, 4, 32);
      rs += __shfl_xor(rs, 8, 32);
      lrun[i] = lrun[i] * f[i] + rs;
    }

    // C-layout P -> LDS (per-wave), rescale accumulators
#pragma unroll
    for (int i = 0; i < 8; ++i)
      Pl[wv][(i + rbase) * PSTR + m] = p[i];
#pragma unroll
    for (int ct = 0; ct < 8; ++ct)
#pragma unroll
      for (int i = 0; i < 8; ++i)
        acc[ct][i] *= f[i];
    __syncthreads();

    // ---- out(16x128) += P(16x16) @ V(16x128) ----
    v2f pa[4];
#pragma unroll
    for (int c = 0; c < 4; ++c)
      pa[c] = *(const v2f*)&Pl[wv][m * PSTR + 4 * c + off];
#pragma unroll
    for (int ct = 0; ct < 8; ++ct) {
#pragma unroll
      for (int c = 0; c < 4; ++c) {
        v2f b;
        b.x = Vl[(4 * c + off)     * KSTR + ct * 16 + m];
        b.y = Vl[(4 * c + off + 1) * KSTR + ct * 16 + m];
        acc[ct] = wmma4(pa[c], b, acc[ct]);
      }
    }
    __syncthreads();   // protect Kl/Vl for next iteration's staging
  }

  // ---- finalize: divide by softmax denominator, store ----
#pragma unroll
  for (int i = 0; i < 8; ++i) {
    float inv  = 1.0f / lrun[i];
    int   head = wv * 16 + rbase + i;
    float* orow =
        out + (size_t)s * (NHEAD * DHEAD) + (size_t)head * DHEAD;
#pragma unroll
    for (int ct = 0; ct < 8; ++ct)
      orow[ct * 16 + m] = acc[ct][i] * inv;
  }
}

// ---------------------------------------------------------------------------
extern "C" void kernel_launch(void* const* d_in, const int* in_sizes, int n_in,
                              void* d_out, int out_size, void* d_ws,
                              size_t ws_size, hipStream_t stream) {
  const float* hidden      = (const float*)d_in[0];
  const float* key_cache   = (const float*)d_in[1];
  const float* value_cache = (const float*)d_in[2];
  const int*   block_tab   = (const int*)d_in[3];
  const int*   ctx_lens    = (const int*)d_in[4];
  const float* Wq          = (const float*)d_in[5];
  const float* bq          = (const float*)d_in[6];
  const float* Wkv         = (const float*)d_in[7];
  const float* bkv         = (const float*)d_in[8];
  const float* Wo          = (const float*)d_in[9];
  const float* bo          = (const float*)d_in[10];
  float* outp = (float*)d_out;

  // Workspace carve-up: q (32*6144) | kv (32*256) | attn (32*6144) = ~1.6 MB
  float* q_buf    = (float*)d_ws;
  float* kv_buf   = q_buf + (size_t)SEQ * HID;
  float* attn_buf = kv_buf + (size_t)SEQ * 256;

  // q = hidden @ Wq + bq            (N = 6144 -> 384 tiles)
  gemm32_wmma<<<dim3(HID / 16), dim3(128), 0, stream>>>(
      hidden, Wq, bq, q_buf, HID);
  // kv = hidden @ Wkv + bkv         (N = 256 -> 16 tiles)
  gemm32_wmma<<<dim3(256 / 16), dim3(128), 0, stream>>>(
      hidden, Wkv, bkv, kv_buf, 256);
  // paged MQA flash attention (incl. self token)
  paged_mqa_wmma<<<dim3(SEQ), dim3(96), 0, stream>>>(
      q_buf, kv_buf, key_cache, value_cache, block_tab, ctx_lens, attn_buf);
  // out = attn @ Wo + bo
  gemm32_wmma<<<dim3(HID / 16), dim3(128), 0, stream>>>(
      attn_buf, Wo, bo, outp, HID);
}